// ProposalModule_9569187135572
// MI455X (gfx1250) — compile-verified
//
#include <hip/hip_runtime.h>

typedef __attribute__((ext_vector_type(2))) float v2f;
typedef __attribute__((ext_vector_type(8))) float v8f;

#define NB 16
#define NP 2048
#define CSEED 256
#define NS 256            // proposals
#define NK 16             // samples per ball
#define RAD 0.3f
#define EPSV 1e-5f
#define CST 272           // padded LDS row stride for CGNL (bank-stagger)

static __device__ __forceinline__ v8f wmma4(v2f a, v2f b, v8f c) {
  // V_WMMA_F32_16X16X4_F32: D = A(16x4) * B(4x16) + C(16x16), wave32
  return __builtin_amdgcn_wmma_f32_16x16x4_f32(false, a, false, b, (short)0, c,
                                               false, false);
}

// ---------------------------------------------------------------- prep ------
struct BnPtrs { const float *beta, *gamma, *mean, *var; };

__global__ __launch_bounds__(128) void prep_kernel(
    BnPtrs b0, BnPtrs b1, BnPtrs b2, BnPtrs b3, BnPtrs b4,
    const float* __restrict__ c1b, const float* __restrict__ c2b,
    float* __restrict__ ss) {
  const int t = threadIdx.x;  // 128 channels
  BnPtrs arr[5] = {b0, b1, b2, b3, b4};
  const float* biases[5] = {nullptr, nullptr, nullptr, c1b, c2b};
  for (int l = 0; l < 5; ++l) {
    float sc = arr[l].gamma[t] * rsqrtf(arr[l].var[t] + EPSV);
    float bias = biases[l] ? biases[l][t] : 0.0f;
    ss[l * 256 + t] = sc;
    ss[l * 256 + 128 + t] = (bias - arr[l].mean[t]) * sc + arr[l].beta[t];
  }
}

// Pad SA W1 (128,259) -> (128,260) with zeros, and fold conv1 weights:
// W1e = conv1_W[:, :128] + conv1_W[:, 128:]   (because x = concat([net,net]))
__global__ __launch_bounds__(256) void padfold_kernel(
    const float* __restrict__ W1sa, const float* __restrict__ c1W,
    float* __restrict__ W1p, float* __restrict__ W1e) {
  const int m = blockIdx.x, t = threadIdx.x;
  if (m < 128) {
    for (int c = t; c < 260; c += 256)
      W1p[m * 260 + c] = (c < 259) ? W1sa[m * 259 + c] : 0.0f;
  } else {
    const int mm = m - 128;
    for (int c = t; c < 128; c += 256)
      W1e[mm * 128 + c] = c1W[mm * 256 + c] + c1W[mm * 256 + 128 + c];
  }
}

// Transpose features (B,C,N) -> (B,N,C) so per-point channel gathers are
// contiguous 1KB rows (coalesced) instead of 256 strided cacheline touches.
__global__ __launch_bounds__(256) void transpose_kernel(
    const float* __restrict__ f, float* __restrict__ ft) {
  __shared__ float tile[32][33];
  const int tx = threadIdx.x & 31, ty = threadIdx.x >> 5;  // 32 x 8
  const int bid = blockIdx.x;
  const int b = bid >> 9;               // 512 tiles per batch
  const int rem = bid & 511;
  const int cb = rem >> 6;              // 8 channel blocks of 32
  const int nb = rem & 63;              // 64 point blocks of 32
  const float* src = f + ((size_t)b * CSEED + cb * 32) * NP + nb * 32;
#pragma unroll
  for (int i = 0; i < 4; ++i) tile[ty + i * 8][tx] = src[(ty + i * 8) * NP + tx];
  __syncthreads();
  float* dst = ft + ((size_t)b * NP + nb * 32) * CSEED + cb * 32;
#pragma unroll
  for (int i = 0; i < 4; ++i) dst[(ty + i * 8) * CSEED + tx] = tile[tx][ty + i * 8];
}

// ----------------------------------------------------------------- fps ------
__global__ __launch_bounds__(256) void fps_kernel(
    const float* __restrict__ xyz,      // (B,N,3)
    float* __restrict__ new_xyz) {      // (B,S,3)
  __shared__ float px[NP], py[NP], pz[NP];
  __shared__ float rv[256];
  __shared__ int ri[256];
  __shared__ int inds[NS];
  __shared__ int s_far;
  const int b = blockIdx.x, t = threadIdx.x;
  const float* base = xyz + (size_t)b * NP * 3;
  for (int i = t; i < NP; i += 256) {
    px[i] = base[i * 3 + 0];
    py[i] = base[i * 3 + 1];
    pz[i] = base[i * 3 + 2];
  }
  if (t == 0) s_far = 0;
  __syncthreads();
  float d[8];
#pragma unroll
  for (int j = 0; j < 8; ++j) d[j] = 1e10f;
  for (int it = 0; it < NS; ++it) {
    const int far = s_far;
    if (t == 0) inds[it] = far;
    const float cx = px[far], cy = py[far], cz = pz[far];
    float bv = -1.0f;
    int bi = 0;
#pragma unroll
    for (int j = 0; j < 8; ++j) {
      const int p = t * 8 + j;
      const float dx = px[p] - cx, dy = py[p] - cy, dz = pz[p] - cz;
      const float nd = fminf(d[j], dx * dx + dy * dy + dz * dz);
      d[j] = nd;
      if (nd > bv) { bv = nd; bi = p; }   // ascending j => first-max kept
    }
    rv[t] = bv; ri[t] = bi;
    __syncthreads();
    for (int s = 128; s > 0; s >>= 1) {
      if (t < s) {
        const float v2 = rv[t + s]; const int i2 = ri[t + s];
        if (v2 > rv[t] || (v2 == rv[t] && i2 < ri[t])) { rv[t] = v2; ri[t] = i2; }
      }
      __syncthreads();
    }
    if (t == 0) s_far = ri[0];
    __syncthreads();
  }
  if (t < NS) {
    const int id = inds[t];
    float* o = new_xyz + ((size_t)b * NS + t) * 3;
    o[0] = px[id]; o[1] = py[id]; o[2] = pz[id];
  }
}

// ----------------------------------------------------------- ball query -----
__global__ __launch_bounds__(256) void ballquery_kernel(
    const float* __restrict__ xyz, const float* __restrict__ new_xyz,
    int* __restrict__ idx) {            // (B,S,16)
  const int wave = threadIdx.x >> 5, lane = threadIdx.x & 31;
  const int gs = blockIdx.x * 8 + wave;    // b*NS + s
  const int b = gs >> 8;
  const float* base = xyz + (size_t)b * NP * 3;
  const float* q = new_xyz + (size_t)gs * 3;
  const float qx = q[0], qy = q[1], qz = q[2];
  int* out = idx + (size_t)gs * NK;
  int count = 0, first = NP - 1;
  const float r2 = RAD * RAD;
  for (int c0 = 0; c0 < NP; c0 += 32) {
    const int p = c0 + lane;
    const float dx = base[p * 3 + 0] - qx, dy = base[p * 3 + 1] - qy,
                dz = base[p * 3 + 2] - qz;
    const bool pred = (dx * dx + dy * dy + dz * dz) < r2;
    const unsigned m = __builtin_amdgcn_ballot_w32(pred);
    if (m) {
      if (count == 0) first = c0 + __builtin_ctz(m);
      const int pos = count + __builtin_popcount(m & ((1u << lane) - 1u));
      if (pred && pos < NK) out[pos] = p;
      count += __builtin_popcount(m);
      if (count >= NK) break;
    }
  }
  count = count < NK ? count : NK;
  if (lane < NK && lane >= count) out[lane] = first;
}

// -------------------------------------------------------- WMMA GEMM tile ----
// One wave computes a 16(out-ch) x 16(col) tile: y = relu(scale*W@X+shift).
// K is always ksteps*4 (weights pre-padded; no guards in the hot loop).
static __device__ __forceinline__ void gemm_tile(
    const float* __restrict__ W, int wstride, int ksteps, const float* Xl,
    float* Yl, const float* __restrict__ scale, const float* __restrict__ shift,
    int m0, int lane) {
  const int mr = lane & 15;
  const int koff = (lane >> 4) << 1;
  const int hof = (lane >> 4) << 3;
  const float* wrow = W + (size_t)(m0 + mr) * wstride;
  __builtin_prefetch(wrow, 0, 1);                 // global_prefetch_b8
  v8f acc = {0.f, 0.f, 0.f, 0.f, 0.f, 0.f, 0.f, 0.f};
  for (int k0 = 0; k0 < ksteps * 4; k0 += 4) {
    const int k = k0 + koff;
    v2f a, bb;
    a.x = wrow[k];
    a.y = wrow[k + 1];
    bb.x = Xl[k * 16 + mr];
    bb.y = Xl[(k + 1) * 16 + mr];
    acc = wmma4(a, bb, acc);
  }
#pragma unroll
  for (int r = 0; r < 8; ++r) {
    const int ch = m0 + r + hof;
    Yl[ch * 16 + mr] = fmaxf(acc[r] * scale[ch] + shift[ch], 0.0f);
  }
}

// -------------------------------------------------------- SA MLP (WMMA) -----
__global__ __launch_bounds__(256) void sa_mlp_kernel(
    const float* __restrict__ xyz, const float* __restrict__ ftr,  // (B,N,C)
    const float* __restrict__ new_xyz, const int* __restrict__ gidx,
    const float* __restrict__ W1p, const float* __restrict__ W2,
    const float* __restrict__ W3, const float* __restrict__ ss,
    float* __restrict__ feat_out) {     // (B,128,S)
  __shared__ float bufA[260 * 16];
  __shared__ float bufB[128 * 16];
  __shared__ int sidx[NK];
  __shared__ float snx[3];
  const int t = threadIdx.x, wave = t >> 5, lane = t & 31;
  const int gs = blockIdx.x;            // b*NS + s
  const int b = gs >> 8;

  if (t < NK) sidx[t] = gidx[(size_t)gs * NK + t];
  if (t < 3) snx[t] = new_xyz[(size_t)gs * 3 + t];
  __syncthreads();
  if (t < 48) {                         // 3 xyz channels
    const int r = t >> 4, k = t & 15;
    bufA[r * 16 + k] =
        (xyz[((size_t)b * NP + sidx[k]) * 3 + r] - snx[r]) * (1.0f / RAD);
  }
  {  // 256 feature channels; transposed layout => coalesced 1KB row reads
#pragma unroll
    for (int k = 0; k < NK; ++k)
      bufA[(3 + t) * 16 + k] = ftr[((size_t)b * NP + sidx[k]) * CSEED + t];
  }
  if (t < 16) bufA[259 * 16 + t] = 0.0f;  // zero pad K: 259 -> 260
  __syncthreads();

  gemm_tile(W1p, 260, 65, bufA, bufB, ss + 0, ss + 128, wave * 16, lane);
  __syncthreads();
  gemm_tile(W2, 128, 32, bufB, bufA, ss + 256, ss + 384, wave * 16, lane);
  __syncthreads();
  {  // layer 3 + max over the 16 samples
    const int mr = lane & 15, koff = (lane >> 4) << 1, hof = (lane >> 4) << 3;
    const int m0 = wave * 16;
    const float* wrow = W3 + (size_t)(m0 + mr) * 128;
    __builtin_prefetch(wrow, 0, 1);
    v8f acc = {0.f, 0.f, 0.f, 0.f, 0.f, 0.f, 0.f, 0.f};
    for (int k0 = 0; k0 < 128; k0 += 4) {
      const int k = k0 + koff;
      v2f a, bb;
      a.x = wrow[k]; a.y = wrow[k + 1];
      bb.x = bufA[k * 16 + mr]; bb.y = bufA[(k + 1) * 16 + mr];
      acc = wmma4(a, bb, acc);
    }
#pragma unroll
    for (int r = 0; r < 8; ++r) {
      const int ch = m0 + r + hof;
      float v = fmaxf(acc[r] * ss[512 + ch] + ss[640 + ch], 0.0f);
#pragma unroll
      for (int off = 1; off < 16; off <<= 1) v = fmaxf(v, __shfl_xor(v, off, 32));
      if (mr == 0) feat_out[((size_t)b * 128 + ch) * NS + (gs & 255)] = v;
    }
  }
}

// ----------------------------------------------------------- CGNL -----------
__global__ __launch_bounds__(256) void cgnl_kernel(
    const float* __restrict__ feat,     // (B,128,256)
    const float* __restrict__ Wt, const float* __restrict__ Wp,
    const float* __restrict__ Wg, const float* __restrict__ Zw,  // (4,32,16)
    const float* __restrict__ gng, const float* __restrict__ gnb,
    float* __restrict__ net) {          // (B,128,256)
  __shared__ float Xc[128 * CST];       // x, later reused for z
  __shared__ float Tm[64 * CST];        // t-matrix
  __shared__ float attp[256 * 4];
  __shared__ float redp[256 * 8];
  __shared__ float attv[4];
  __shared__ float stat[8];             // mu[4], rinv[4]
  const int t = threadIdx.x, wave = t >> 5, lane = t & 31;
  const int b = blockIdx.x;
  const float* xb = feat + (size_t)b * 128 * 256;
  for (int e = t; e < 128 * 256; e += 256)
    Xc[(e >> 8) * CST + (e & 255)] = xb[e];
  __syncthreads();

  const int mr = lane & 15, koff = (lane >> 4) << 1, hof = (lane >> 4) << 3;
  // phase 1: t into LDS, att = sum(q*g) partials per group
  float ap[4] = {0.f, 0.f, 0.f, 0.f};
  for (int tile = wave; tile < 64; tile += 8) {
    const int rt = tile & 3, ct = tile >> 2;
    const int m = rt * 16 + mr, n = ct * 16 + mr;
    v8f acc = {0.f, 0.f, 0.f, 0.f, 0.f, 0.f, 0.f, 0.f};
    for (int k0 = 0; k0 < 128; k0 += 4) {
      const int k = k0 + koff;
      v2f a, bb;
      a.x = Wt[m * 128 + k]; a.y = Wt[m * 128 + k + 1];
      bb.x = Xc[k * CST + n]; bb.y = Xc[(k + 1) * CST + n];
      acc = wmma4(a, bb, acc);
    }
#pragma unroll
    for (int r = 0; r < 8; ++r) Tm[(rt * 16 + r + hof) * CST + n] = acc[r];
    v8f aq = {0.f, 0.f, 0.f, 0.f, 0.f, 0.f, 0.f, 0.f};
    v8f ag = {0.f, 0.f, 0.f, 0.f, 0.f, 0.f, 0.f, 0.f};
    for (int k0 = 0; k0 < 128; k0 += 4) {
      const int k = k0 + koff;
      v2f a1, a2, bb;
      a1.x = Wp[m * 128 + k]; a1.y = Wp[m * 128 + k + 1];
      a2.x = Wg[m * 128 + k]; a2.y = Wg[m * 128 + k + 1];
      bb.x = Xc[k * CST + n]; bb.y = Xc[(k + 1) * CST + n];
      aq = wmma4(a1, bb, aq);
      ag = wmma4(a2, bb, ag);
    }
    float p = 0.f;
#pragma unroll
    for (int r = 0; r < 8; ++r) p += aq[r] * ag[r];
    ap[rt] += p;
  }
#pragma unroll
  for (int g = 0; g < 4; ++g) attp[t * 4 + g] = ap[g];
  __syncthreads();
  if (t < 4) {  // deterministic serial reduction
    float s = 0.f;
    for (int i = 0; i < 256; ++i) s += attp[i * 4 + t];
    attv[t] = s;
  }
  __syncthreads();

  // phase 2: z = att[g] * (Zw_g @ t_g), stored over Xc; group sum/sumsq
  float sump[4] = {0.f, 0.f, 0.f, 0.f}, sqp[4] = {0.f, 0.f, 0.f, 0.f};
  for (int tile = wave; tile < 128; tile += 8) {
    const int g = tile >> 5, rem = tile & 31, mt = rem & 1, ct = rem >> 1;
    const int n = ct * 16 + mr, m = mt * 16 + mr;
    const float* zrow = Zw + (size_t)(g * 32 + m) * 16;
    v8f acc = {0.f, 0.f, 0.f, 0.f, 0.f, 0.f, 0.f, 0.f};
    for (int k0 = 0; k0 < 16; k0 += 4) {
      const int k = k0 + koff;
      v2f a, bb;
      a.x = zrow[k]; a.y = zrow[k + 1];
      bb.x = Tm[(g * 16 + k) * CST + n]; bb.y = Tm[(g * 16 + k + 1) * CST + n];
      acc = wmma4(a, bb, acc);
    }
    const float av = attv[g];
#pragma unroll
    for (int r = 0; r < 8; ++r) {
      const int ch = g * 32 + mt * 16 + r + hof;
      const float v = acc[r] * av;
      Xc[ch * CST + n] = v;
      sump[g] += v;
      sqp[g] += v * v;
    }
  }
#pragma unroll
  for (int g = 0; g < 4; ++g) { redp[t * 8 + g] = sump[g]; redp[t * 8 + 4 + g] = sqp[g]; }
  __syncthreads();
  if (t < 4) {
    float s = 0.f, q = 0.f;
    for (int i = 0; i < 256; ++i) { s += redp[i * 8 + t]; q += redp[i * 8 + 4 + t]; }
    const float mu = s * (1.0f / 8192.0f);
    stat[t] = mu;
    stat[4 + t] = rsqrtf(q * (1.0f / 8192.0f) - mu * mu + EPSV);
  }
  __syncthreads();
  // phase 3: groupnorm + residual
  for (int e = t; e < 128 * 256; e += 256) {
    const int ch = e >> 8, n = e & 255, g = ch >> 5;
    float v = (Xc[ch * CST + n] - stat[g]) * stat[4 + g];
    net[(size_t)b * 128 * 256 + e] = v * gng[ch] + gnb[ch] + xb[e];
  }
}

// ------------------------------------------------------ prediction head -----
__global__ __launch_bounds__(256) void head_kernel(
    const float* __restrict__ net, const float* __restrict__ W1e,
    const float* __restrict__ W2, const float* __restrict__ W3,
    const float* __restrict__ c3b, const float* __restrict__ ss,
    const float* __restrict__ new_xyz, float* __restrict__ out) {  // (B,256,21)
  __shared__ float Xa[128 * 16];
  __shared__ float Xb[128 * 16];
  const int t = threadIdx.x, wave = t >> 5, lane = t & 31;
  const int b = blockIdx.x >> 4, ct = blockIdx.x & 15;
  for (int e = t; e < 2048; e += 256) {
    const int ch = e >> 4, n = e & 15;
    Xa[ch * 16 + n] = net[((size_t)b * 128 + ch) * 256 + ct * 16 + n];
  }
  __syncthreads();
  gemm_tile(W1e, 128, 32, Xa, Xb, ss + 768, ss + 896, wave * 16, lane);
  __syncthreads();
  gemm_tile(W2, 128, 32, Xb, Xa, ss + 1024, ss + 1152, wave * 16, lane);
  __syncthreads();
  const int mr = lane & 15, koff = (lane >> 4) << 1, hof = (lane >> 4) << 3;
  if (wave < 2) {  // conv3 (21 rows) + transpose write-out
    const int m0 = wave * 16;
    const int m = m0 + mr;
    const float* wrow = W3 + (size_t)m * 128;
    v8f acc = {0.f, 0.f, 0.f, 0.f, 0.f, 0.f, 0.f, 0.f};
    for (int k0 = 0; k0 < 128; k0 += 4) {
      const int k = k0 + koff;
      v2f a, bb;
      a.x = (m < 21) ? wrow[k] : 0.0f;
      a.y = (m < 21) ? wrow[k + 1] : 0.0f;
      bb.x = Xa[k * 16 + mr]; bb.y = Xa[(k + 1) * 16 + mr];
      acc = wmma4(a, bb, acc);
    }
    const int nG = ct * 16 + mr;
#pragma unroll
    for (int r = 0; r < 8; ++r) {
      const int co = m0 + r + hof;
      if (co < 21) {
        float v = acc[r] + c3b[co];
        if (co >= 2 && co < 5) v += new_xyz[((size_t)b * 256 + nG) * 3 + (co - 2)];
        out[((size_t)b * 256 + nG) * 21 + co] = v;
      }
    }
  }
}

// -------------------------------------------------------------- launch ------
extern "C" void kernel_launch(void* const* d_in, const int* in_sizes, int n_in,
                              void* d_out, int out_size, void* d_ws,
                              size_t ws_size, hipStream_t stream) {
  (void)in_sizes; (void)n_in; (void)out_size; (void)ws_size;
  // pytree leaf order (dict keys sorted; ASCII: 'W' < lowercase)
  const float* xyz_o  = (const float*)d_in[0];
  const float* feat_o = (const float*)d_in[1];
  // d_in[2], d_in[3]: part branch -- dead code in the reference, skipped.
  BnPtrs bn1  = {(const float*)d_in[4],  (const float*)d_in[5],
                 (const float*)d_in[6],  (const float*)d_in[7]};
  BnPtrs bn2  = {(const float*)d_in[8],  (const float*)d_in[9],
                 (const float*)d_in[10], (const float*)d_in[11]};
  const float* cg_g  = (const float*)d_in[12];
  const float* cg_gb = (const float*)d_in[13];
  const float* cg_gg = (const float*)d_in[14];
  const float* cg_p  = (const float*)d_in[15];
  const float* cg_t  = (const float*)d_in[16];
  const float* cg_z  = (const float*)d_in[17];
  // d_in[18..23]: cgnl_part -- dead.
  const float* c1W = (const float*)d_in[24];
  const float* c1b = (const float*)d_in[25];
  const float* c2W = (const float*)d_in[26];
  const float* c2b = (const float*)d_in[27];
  const float* c3W = (const float*)d_in[28];
  const float* c3b = (const float*)d_in[29];
  const float* saW1 = (const float*)d_in[30];
  BnPtrs sb1 = {(const float*)d_in[31], (const float*)d_in[32],
                (const float*)d_in[33], (const float*)d_in[34]};
  const float* saW2 = (const float*)d_in[35];
  BnPtrs sb2 = {(const float*)d_in[36], (const float*)d_in[37],
                (const float*)d_in[38], (const float*)d_in[39]};
  const float* saW3 = (const float*)d_in[40];
  BnPtrs sb3 = {(const float*)d_in[41], (const float*)d_in[42],
                (const float*)d_in[43], (const float*)d_in[44]};

  float* ws = (float*)d_ws;
  float* new_xyz = ws;                                  // 16*256*3    = 12288
  int*   idx     = (int*)(ws + 12288);                  // 16*256*16   = 65536
  float* featbuf = ws + 12288 + 65536;                  // 16*128*256  = 524288
  float* netbuf  = featbuf + 524288;                    // 16*128*256  = 524288
  float* ss      = netbuf + 524288;                     // 5*256       = 1280
  float* W1p     = ss + 1280;                           // 128*260     = 33280
  float* W1e     = W1p + 33280;                         // 128*128     = 16384
  float* ftr     = W1e + 16384;                         // 16*2048*256 = 8388608
  float* outp    = (float*)d_out;

  prep_kernel<<<1, 128, 0, stream>>>(sb1, sb2, sb3, bn1, bn2, c1b, c2b, ss);
  padfold_kernel<<<256, 256, 0, stream>>>(saW1, c1W, W1p, W1e);
  transpose_kernel<<<NB * 512, 256, 0, stream>>>(feat_o, ftr);
  fps_kernel<<<NB, 256, 0, stream>>>(xyz_o, new_xyz);
  ballquery_kernel<<<NB * NS / 8, 256, 0, stream>>>(xyz_o, new_xyz, idx);
  sa_mlp_kernel<<<NB * NS, 256, 0, stream>>>(xyz_o, ftr, new_xyz, idx, W1p,
                                             saW2, saW3, ss, featbuf);
  cgnl_kernel<<<NB, 256, 0, stream>>>(featbuf, cg_t, cg_p, cg_g, cg_z, cg_gg,
                                      cg_gb, netbuf);
  head_kernel<<<NB * 16, 256, 0, stream>>>(netbuf, W1e, c2W, c3W, c3b, ss,
                                           new_xyz, outp);
}